// UnifyModel_35424890257740
// MI455X (gfx1250) — compile-verified
//
#include <hip/hip_runtime.h>
#include <hip/hip_bf16.h>

typedef __attribute__((ext_vector_type(2))) float v2f;
typedef __attribute__((ext_vector_type(8))) float v8f;

#define IN_CH   128
// ---------------------------------------------------------------------------
// f32 atomic add -> hardware global_atomic_add_f32
// ---------------------------------------------------------------------------
__device__ __forceinline__ void atomAddF(float* p, float v) {
#if defined(__AMDGCN__)
    unsafeAtomicAdd(p, v);
#else
    atomicAdd(p, v);
#endif
}

// ---------------------------------------------------------------------------
// Degree / normalization kernels
// ---------------------------------------------------------------------------
__global__ void deg_init_kernel(float* __restrict__ deg, int n) {
    int i = blockIdx.x * blockDim.x + threadIdx.x;
    if (i < n) deg[i] = 1.0f;               // self-loop contributes 1
}

__global__ void deg_count_kernel(const long long* __restrict__ ei,
                                 float* __restrict__ deg, int nE) {
    int e = blockIdx.x * blockDim.x + threadIdx.x;
    if (e < nE) {
        long long d = ei[nE + e];           // dst row of edge_index
        atomAddF(&deg[d], 1.0f);
    }
}

__global__ void dinv_kernel(float* __restrict__ deg_dinv, int n) {
    int i = blockIdx.x * blockDim.x + threadIdx.x;
    if (i < n) {
        float dg = deg_dinv[i];             // >= 1 always (self-loop)
        deg_dinv[i] = 1.0f / sqrtf(dg);
    }
}

// ---------------------------------------------------------------------------
// GEMM  Y[M,N] = X[M,128] @ W[128,N]  via V_WMMA_F32_16X16X4_F32.
// blockDim.x = 32 * (N/16); wave w owns the 16x16 output tile at col w*16.
// Fragment layouts per CDNA5 ISA 7.12.2:
//   A 16x4 f32 : lane m = lane&15, VGPR pair holds K = kb, kb+1 with
//                kb = (lane>=16 ? 2 : 0)
//   B 4x16  f32: lane n = lane&15, same kb split across lane halves
//   C/D 16x16  : VGPR r -> M = r + (lane>=16 ? 8 : 0), N = lane&15
// ---------------------------------------------------------------------------
template<int N>
__global__ void gemm_wmma_f32(const float* __restrict__ X,
                              const float* __restrict__ W,
                              float* __restrict__ Y, int M) {
    const int lane    = threadIdx.x & 31;
    const int wv      = threadIdx.x >> 5;          // column tile index
    const int rowBase = blockIdx.x << 4;           // 16 rows per block
    const int mloc    = lane & 15;
    const int kb      = (lane >> 4) << 1;          // 0 or 2
    const int n       = (wv << 4) + (lane & 15);   // global output column

    int arow = rowBase + mloc;
    if (arow > M - 1) arow = M - 1;                // clamp (M divisible by 16 here)
    const float* __restrict__ xrow = X + (size_t)arow * IN_CH + kb;

    v8f c = {};
    #pragma unroll
    for (int k = 0; k < IN_CH; k += 4) {
        v2f a = *(const v2f*)(xrow + k);           // K = k+kb, k+kb+1 (contiguous)
        v2f b;
        b.x = W[(size_t)(k + kb)     * N + n];
        b.y = W[(size_t)(k + kb + 1) * N + n];
        c = __builtin_amdgcn_wmma_f32_16x16x4_f32(
                /*neg_a=*/false, a, /*neg_b=*/false, b,
                /*c_mod=*/(short)0, c, /*reuse_a=*/false, /*reuse_b=*/false);
    }

    const int rowHi = (lane >> 4) << 3;            // 0 or 8
    #pragma unroll
    for (int r = 0; r < 8; ++r) {
        int row = rowBase + rowHi + r;
        if (row < M) Y[(size_t)row * N + n] = c[r];
    }
}

// ---------------------------------------------------------------------------
// agg[i,c] = dinv[i]^2 * xw[i,c]   (self-loop term; also initializes agg)
// ---------------------------------------------------------------------------
template<int C>
__global__ void self_init_kernel(const float* __restrict__ xw,
                                 const float* __restrict__ dinv,
                                 float* __restrict__ agg, int total) {
    int idx = blockIdx.x * blockDim.x + threadIdx.x;
    if (idx < total) {
        int i    = idx >> (C == 128 ? 7 : 6);
        float di = dinv[i];
        agg[idx] = di * di * xw[idx];
    }
}

// ---------------------------------------------------------------------------
// Edge scatter: one wave per edge.  agg[dst] += dinv[s]*dinv[d] * xw[src]
// C=128 -> float4 per lane (4 atomics); C=64 -> float2 per lane (2 atomics)
// ---------------------------------------------------------------------------
template<int C>
__global__ void scatter_kernel(const long long* __restrict__ ei,
                               const float* __restrict__ dinv,
                               const float* __restrict__ xw,
                               float* __restrict__ agg, int nE) {
    int gid  = blockIdx.x * blockDim.x + threadIdx.x;
    int e    = gid >> 5;
    int lane = gid & 31;
    if (e >= nE) return;

    long long s = ei[e];
    long long d = ei[nE + e];
    float w = dinv[s] * dinv[d];

    if (C == 128) {
        const float4* src = (const float4*)(xw + (size_t)s * C) + lane;
        float*        dst = agg + (size_t)d * C + lane * 4;
        float4 v = *src;
        atomAddF(dst + 0, w * v.x);
        atomAddF(dst + 1, w * v.y);
        atomAddF(dst + 2, w * v.z);
        atomAddF(dst + 3, w * v.w);
    } else {
        const float2* src = (const float2*)(xw + (size_t)s * C) + lane;
        float*        dst = agg + (size_t)d * C + lane * 2;
        float2 v = *src;
        atomAddF(dst + 0, w * v.x);
        atomAddF(dst + 1, w * v.y);
    }
}

// ---------------------------------------------------------------------------
// out[i,c] = relu(agg[i,c] + b[c])
// ---------------------------------------------------------------------------
template<int C>
__global__ void bias_relu_kernel(const float* __restrict__ agg,
                                 const float* __restrict__ b,
                                 float* __restrict__ out, int total) {
    int idx = blockIdx.x * blockDim.x + threadIdx.x;
    if (idx < total) {
        float v = agg[idx] + b[idx & (C - 1)];
        out[idx] = v > 0.0f ? v : 0.0f;
    }
}

// ---------------------------------------------------------------------------
// Launch: 2-layer GCN
//   inputs: x[N,128] f32, edge_index[2,E] i64, W1[128,128], b1[128],
//           W2[128,64], b2[64];  output: [N,64] f32
// ---------------------------------------------------------------------------
extern "C" void kernel_launch(void* const* d_in, const int* in_sizes, int n_in,
                              void* d_out, int out_size, void* d_ws, size_t ws_size,
                              hipStream_t stream) {
    const float*     x    = (const float*)d_in[0];
    const long long* ei   = (const long long*)d_in[1];
    const float*     W1   = (const float*)d_in[2];
    const float*     b1   = (const float*)d_in[3];
    const float*     W2   = (const float*)d_in[4];
    const float*     b2   = (const float*)d_in[5];
    float*           out  = (float*)d_out;

    const int nNodes = in_sizes[0] / 128;       // 50000
    const int nEdges = in_sizes[1] / 2;         // 800000
    const int H = 128, O = 64;

    // Workspace layout (floats):
    //   [0, nNodes)                       : deg -> dinv (in place)
    //   buf1 [nNodes*128]                 : xw1, later h
    //   buf2 [nNodes*128]                 : agg1; later xw2 (lo half) + agg2 (hi half)
    float* fws  = (float*)d_ws;
    float* dinv = fws;
    float* buf1 = fws + (size_t)nNodes;                    // 200000 B offset, 16B aligned
    float* buf2 = buf1 + (size_t)nNodes * H;
    float* xw2  = buf2;
    float* agg2 = buf2 + (size_t)nNodes * O;

    const int T = 256;
    dim3 blk(T);

    // --- degree / normalization (shared by both layers) ---
    deg_init_kernel<<<(nNodes + T - 1) / T, blk, 0, stream>>>(dinv, nNodes);
    deg_count_kernel<<<(nEdges + T - 1) / T, blk, 0, stream>>>(ei, dinv, nEdges);
    dinv_kernel<<<(nNodes + T - 1) / T, blk, 0, stream>>>(dinv, nNodes);

    const int rowTiles    = (nNodes + 15) / 16;            // 3125
    const int edgeThreads = nEdges * 32;                   // one wave per edge
    const int edgeBlocks  = (edgeThreads + T - 1) / T;

    // --- layer 1: 128 -> 128 ---
    {
        const int total = nNodes * H;
        gemm_wmma_f32<128><<<rowTiles, 256, 0, stream>>>(x, W1, buf1, nNodes);
        self_init_kernel<128><<<(total + T - 1) / T, blk, 0, stream>>>(buf1, dinv, buf2, total);
        scatter_kernel<128><<<edgeBlocks, blk, 0, stream>>>(ei, dinv, buf1, buf2, nEdges);
        bias_relu_kernel<128><<<(total + T - 1) / T, blk, 0, stream>>>(buf2, b1, buf1, total);
    }

    // --- layer 2: 128 -> 64 ---
    {
        const int total = nNodes * O;
        gemm_wmma_f32<64><<<rowTiles, 128, 0, stream>>>(buf1, W2, xw2, nNodes);
        self_init_kernel<64><<<(total + T - 1) / T, blk, 0, stream>>>(xw2, dinv, agg2, total);
        scatter_kernel<64><<<edgeBlocks, blk, 0, stream>>>(ei, dinv, xw2, agg2, nEdges);
        bias_relu_kernel<64><<<(total + T - 1) / T, blk, 0, stream>>>(agg2, b2, out, total);
    }
}